// ConsistencyLoss_27711128994500
// MI455X (gfx1250) — compile-verified
//
#include <hip/hip_runtime.h>

// ---------------------------------------------------------------------------
// Problem constants (from setup_inputs):
//   input   : [8, 64, 256, 256]  f32
//   feature : [8, 256, 64, 64]   f32
//   sp      : [8, 1, 256, 256]   i32 in [0,256)
//   num     : 256
// Output: scalar f32.
//
// Math reduction: per-segment mean is a SCALAR over (channels x pixels), and
// bilinear resize (align_corners) commutes with the channel sum, so we need
// only per-pixel channel sums, a 64x64 -> 256x256 bilinear upsample of the
// feature sum map, segment sums, and a 256x256 pairwise reduction per batch.
// Total HBM traffic ~170 MB -> ~7.3 us at 23.3 TB/s (memory-bound).
// ---------------------------------------------------------------------------

typedef __attribute__((ext_vector_type(2))) float v2f;
typedef __attribute__((ext_vector_type(8))) float v8f;

#define BATCH   8
#define C_IN    64
#define C_FEAT  256
#define HW_IN   65536   // 256*256
#define W_IN    256
#define HW_F    4096    // 64*64
#define W_F     64
#define NSEG    256

// workspace layout (floats)
#define OFF_XS    0                         // BATCH*HW_IN   = 524288
#define OFF_FS    (BATCH * HW_IN)           // BATCH*HW_F    =  32768
#define OFF_SEGI  (OFF_FS + BATCH * HW_F)   // BATCH*NSEG    =   2048
#define OFF_SEGF  (OFF_SEGI + BATCH * NSEG)
#define OFF_CNT   (OFF_SEGF + BATCH * NSEG)
#define WS_FLOATS (OFF_CNT + BATCH * NSEG)  // ~563K floats ~ 2.2 MB

// ---------------------------------------------------------------------------
// Kernel 0: zero the segment accumulators and the output scalar.
// ---------------------------------------------------------------------------
__global__ void cl_init_kernel(float* seg_base, float* out) {
    int i = blockIdx.x * blockDim.x + threadIdx.x;
    if (i < 3 * BATCH * NSEG) seg_base[i] = 0.0f;
    if (i == 0) out[0] = 0.0f;
}

// ---------------------------------------------------------------------------
// Kernels 1/2: channel sum via V_WMMA_F32_16X16X4_F32 with a ones A-matrix.
//   D = ones(16x4) x B(4x16) + C  =>  every row of D is the column sum of B.
// Wave handles 16 consecutive pixels (columns). Per K-step of 4 channels:
//   lane L, comp j loads x[channel = c + 2*(L>>4) + j, pixel = base+(L&15)]
// With A==1 the accumulated sum is invariant to the exact (lane,comp)->K
// permutation the HW uses; only the column = lane%16 mapping matters.
// Each element is loaded exactly once; lanes 0-15 / 16-31 each form one
// contiguous 64-byte segment per load instruction (coalesced).
//
// Template constants => channel strides are compile-time: for the feature
// tensor (16 KB stride) every channel offset fits the 24-bit IOFFSET, so the
// fully-unrolled loop becomes immediate-offset loads off one base register;
// for the input tensor (256 KB stride) two base registers cover all 64 rows.
// Block = 256 threads = 8 waves = 128 pixels.
// ---------------------------------------------------------------------------
template <int C, int HW, int TILES>
__global__ void cl_chansum_wmma(const float* __restrict__ x,
                                float* __restrict__ out) {
    const int wave = threadIdx.x >> 5;
    const int lane = threadIdx.x & 31;
    const int tile = blockIdx.x % TILES;   // TILES is a power of two
    const int b    = blockIdx.x / TILES;
    const int pix  = tile * 128 + wave * 16 + (lane & 15);
    const int kq   = (lane >> 4) << 1;     // 0 or 2

    const float* base = x + (size_t)b * C * HW + (size_t)kq * HW + pix;

    v2f a;  a.x = 1.0f; a.y = 1.0f;        // ones A-matrix
    v8f acc = {0.f, 0.f, 0.f, 0.f, 0.f, 0.f, 0.f, 0.f};

#pragma unroll
    for (int c = 0; c < C; c += 4) {
        v2f bm;
        bm.x = base[(size_t)c * HW];
        bm.y = base[(size_t)c * HW + HW];
        acc = __builtin_amdgcn_wmma_f32_16x16x4_f32(
            /*neg_a=*/false, a, /*neg_b=*/false, bm,
            /*c_mod=*/(short)0, acc, /*reuse_a=*/false, /*reuse_b=*/false);
    }
    // All D rows are identical; D vgpr0 on lane L = colsum[L & 15].
    if (lane < 16) out[(size_t)b * HW + pix] = acc[0];
}

// ---------------------------------------------------------------------------
// Kernel 3: fused bilinear upsample (64->256, align_corners) of the feature
// channel-sum map + per-segment accumulation of both maps and counts.
// LDS histogram per block (ds_add_f32), then one global atomic per bin.
// 64 blocks per batch, 1024 pixels per block, 4 pixels per thread.
// ---------------------------------------------------------------------------
__global__ void cl_segment_kernel(const float* __restrict__ xs,
                                  const float* __restrict__ fs,
                                  const int* __restrict__ sp,
                                  float* __restrict__ segI,
                                  float* __restrict__ segF,
                                  float* __restrict__ cnt) {
    __shared__ float sI[NSEG], sF[NSEG], sC[NSEG];
    const int b   = blockIdx.x >> 6;   // 64 blocks per batch
    const int blk = blockIdx.x & 63;
    const int t   = threadIdx.x;       // blockDim.x == 256 == NSEG

    sI[t] = 0.0f; sF[t] = 0.0f; sC[t] = 0.0f;
    __syncthreads();

    const float* f = fs + (size_t)b * HW_F;
    const int pbase = blk * 1024;
    const float scale = 63.0f / 255.0f;   // align_corners: (in-1)/(out-1)

#pragma unroll
    for (int i = 0; i < 4; ++i) {
        const int p = pbase + i * 256 + t;
        const int y = p >> 8;
        const int x = p & 255;

        const float fy = (float)y * scale;
        const float fx = (float)x * scale;
        const int y0 = (int)fy;                 // fy >= 0 -> trunc == floor
        const int x0 = (int)fx;
        const float ty = fy - (float)y0;
        const float tx = fx - (float)x0;
        const int y1 = min(y0 + 1, W_F - 1);
        const int x1 = min(x0 + 1, W_F - 1);

        const float v00 = f[y0 * W_F + x0];
        const float v01 = f[y0 * W_F + x1];
        const float v10 = f[y1 * W_F + x0];
        const float v11 = f[y1 * W_F + x1];
        const float up = (1.0f - ty) * ((1.0f - tx) * v00 + tx * v01)
                       +         ty  * ((1.0f - tx) * v10 + tx * v11);

        const int   id = sp[(size_t)b * HW_IN + p];
        const float xv = xs[(size_t)b * HW_IN + p];

        atomicAdd(&sI[id], xv);
        atomicAdd(&sF[id], up);
        atomicAdd(&sC[id], 1.0f);
    }
    __syncthreads();

    atomicAdd(&segI[b * NSEG + t], sI[t]);
    atomicAdd(&segF[b * NSEG + t], sF[t]);
    atomicAdd(&cnt [b * NSEG + t], sC[t]);
}

// ---------------------------------------------------------------------------
// Kernel 4: per-batch pairwise reduction.
//   m1[s] = segI/(cnt*64); m2[s] = segF/(cnt*256)
//   result += sum_{i,j} | |m1_i-m1_j| - |m2_i-m2_j| | / (B*NSEG*NSEG)
// (mat1-mat2 = (1-|d1|)-(1-|d2|) = |d2|-|d1|, abs is symmetric.)
// One block per batch, 256 threads, LDS tree reduction, atomicAdd to out.
// ---------------------------------------------------------------------------
__global__ void cl_pairwise_kernel(const float* __restrict__ segI,
                                   const float* __restrict__ segF,
                                   const float* __restrict__ cnt,
                                   float* __restrict__ out) {
    __shared__ float m1[NSEG], m2[NSEG], red[NSEG];
    const int b = blockIdx.x;
    const int t = threadIdx.x;

    const float c = cnt[b * NSEG + t];
    m1[t] = segI[b * NSEG + t] / (c * (float)C_IN);
    m2[t] = segF[b * NSEG + t] / (c * (float)C_FEAT);
    __syncthreads();

    const float a1 = m1[t];
    const float a2 = m2[t];
    float s = 0.0f;
#pragma unroll 8
    for (int j = 0; j < NSEG; ++j)
        s += fabsf(fabsf(a1 - m1[j]) - fabsf(a2 - m2[j]));

    red[t] = s;
    __syncthreads();
    for (int off = 128; off > 0; off >>= 1) {
        if (t < off) red[t] += red[t + off];
        __syncthreads();
    }
    if (t == 0)
        atomicAdd(out, red[0] * (1.0f / ((float)BATCH * NSEG * NSEG)));
}

// ---------------------------------------------------------------------------
extern "C" void kernel_launch(void* const* d_in, const int* in_sizes, int n_in,
                              void* d_out, int out_size, void* d_ws, size_t ws_size,
                              hipStream_t stream) {
    const float* input   = (const float*)d_in[0];   // [8,64,256,256]
    const float* feature = (const float*)d_in[1];   // [8,256,64,64]
    const int*   sp      = (const int*)  d_in[2];   // [8,1,256,256]
    (void)in_sizes; (void)n_in; (void)out_size; (void)ws_size;

    float* ws   = (float*)d_ws;
    float* xs   = ws + OFF_XS;
    float* fs   = ws + OFF_FS;
    float* segI = ws + OFF_SEGI;
    float* segF = ws + OFF_SEGF;
    float* cnt  = ws + OFF_CNT;
    float* out  = (float*)d_out;

    // 0) zero accumulators + output scalar
    cl_init_kernel<<<(3 * BATCH * NSEG + 255) / 256, 256, 0, stream>>>(segI, out);

    // 1) input channel sum: 512 tiles-of-128px per batch -> 4096 blocks
    cl_chansum_wmma<C_IN, HW_IN, HW_IN / 128>
        <<<BATCH * (HW_IN / 128), 256, 0, stream>>>(input, xs);

    // 2) feature channel sum: 32 tiles per batch -> 256 blocks
    cl_chansum_wmma<C_FEAT, HW_F, HW_F / 128>
        <<<BATCH * (HW_F / 128), 256, 0, stream>>>(feature, fs);

    // 3) upsample + segment histogram: 64 blocks per batch
    cl_segment_kernel<<<BATCH * 64, 256, 0, stream>>>(xs, fs, sp, segI, segF, cnt);

    // 4) pairwise reduction: one block per batch
    cl_pairwise_kernel<<<BATCH, 256, 0, stream>>>(segI, segF, cnt, out);
}